// Occlusion_82377472737647
// MI455X (gfx1250) — compile-verified
//
#include <hip/hip_runtime.h>
#include <cstdint>

// Problem geometry from the reference: B=8, C=2, H=W=1024 (H,W hardcoded; B derived).
#define H_DIM 1024
#define W_DIM 1024
#define PLANE (H_DIM * W_DIM)
#define BLOCK 256
#define PIXPT 2   // pixels per thread; one block = 512 contiguous pixels = half a row

typedef __attribute__((ext_vector_type(2))) float v2f;

// Bilinear grid-sample of (b0p,b1p) at (y+f0, x+f1) with torch grid_sample
// defaults (align_corners=False, zero padding), replicating the reference's
// fp32 op order exactly, plus the occlusion test.
__device__ __forceinline__ void sample_one(
    const float* __restrict__ b0p, const float* __restrict__ b1p,
    int y, int x, float f0, float f1,
    float& r0, float& r1, float& occ)
{
    float ii = (float)y + f0;
    float jj = (float)x + f1;
    ii = fminf(fmaxf(ii, 0.0f), (float)(H_DIM - 1));
    jj = fminf(fmaxf(jj, 0.0f), (float)(W_DIM - 1));
    const float gy = 2.0f * ii / (float)(H_DIM - 1) - 1.0f;
    const float gx = 2.0f * jj / (float)(W_DIM - 1) - 1.0f;
    const float sx = ((gx + 1.0f) * (float)W_DIM - 1.0f) * 0.5f;
    const float sy = ((gy + 1.0f) * (float)H_DIM - 1.0f) * 0.5f;

    const float x0f = floorf(sx), y0f = floorf(sy);
    const float wx1 = sx - x0f, wx0 = 1.0f - wx1;
    const float wy1 = sy - y0f, wy0 = 1.0f - wy1;
    const int x0 = (int)x0f, y0 = (int)y0f;
    const int x1 = x0 + 1,  y1 = y0 + 1;

    // zero-padding validity folded into tap weights
    const float vx0 = (x0 >= 0 && x0 < W_DIM) ? 1.0f : 0.0f;
    const float vx1 = (x1 >= 0 && x1 < W_DIM) ? 1.0f : 0.0f;
    const float vy0 = (y0 >= 0 && y0 < H_DIM) ? 1.0f : 0.0f;
    const float vy1 = (y1 >= 0 && y1 < H_DIM) ? 1.0f : 0.0f;
    const float w00 = wy0 * wx0 * vy0 * vx0;
    const float w01 = wy0 * wx1 * vy0 * vx1;
    const float w10 = wy1 * wx0 * vy1 * vx0;
    const float w11 = wy1 * wx1 * vy1 * vx1;

    const int x0c = min(max(x0, 0), W_DIM - 1);
    const int x1c = min(max(x1, 0), W_DIM - 1);
    const int y0c = min(max(y0, 0), H_DIM - 1);
    const int y1c = min(max(y1, 0), H_DIM - 1);
    const int o00 = y0c * W_DIM + x0c;
    const int o01 = y0c * W_DIM + x1c;
    const int o10 = y1c * W_DIM + x0c;
    const int o11 = y1c * W_DIM + x1c;

    r0 = w00 * b0p[o00] + w01 * b0p[o01] + w10 * b0p[o10] + w11 * b0p[o11];
    r1 = w00 * b1p[o00] + w01 * b1p[o01] + w10 * b1p[o10] + w11 * b1p[o11];

    const float as = fabsf(f0 + r0) + fabsf(f1 + r1);
    const float af = fabsf(f0) + fabsf(f1);
    const float ar = fabsf(r0) + fabsf(r1);
    occ = ((as * as) >= 0.01f * (af * af + ar * ar) + 0.05f) ? 1.0f : 0.0f;
}

__global__ __launch_bounds__(BLOCK) void occlusion_kernel(
    const float* __restrict__ dp_f,   // [B,2,H,W]
    const float* __restrict__ dp_b,   // [B,2,H,W]
    float* __restrict__ occ_out,      // [B,H,W]   (bool as 0.0/1.0)
    float* __restrict__ rev_out,      // [B,2,H,W]
    int npix)                         // B*H*W (multiple of BLOCK*PIXPT)
{
    const int tid      = threadIdx.x;
    const int blockPix = blockIdx.x * (BLOCK * PIXPT);   // block tile start (uniform)
    const int idx0     = blockPix + tid * PIXPT;         // first pixel of this thread
    if (idx0 >= npix) return;

    // Block tile = 512 contiguous pixels => same batch and same row for the
    // whole block: b and y derive from blockIdx only and live in SGPRs.
    const int b   = blockPix >> 20;                      // / PLANE (2^20), uniform
    const int y   = (blockPix & (PLANE - 1)) >> 10;      // row, uniform
    const int rem = idx0 & (PLANE - 1);                  // per-thread (even)
    const int x   = rem & (W_DIM - 1);                   // per-thread column (even)

    const float* __restrict__ f0p = dp_f + (size_t)b * 2 * PLANE;  // fwd flow ch0
    const float* __restrict__ f1p = f0p + PLANE;                   // fwd flow ch1
    const float* __restrict__ b0p = dp_b + (size_t)b * 2 * PLANE;  // bwd flow ch0
    const float* __restrict__ b1p = b0p + PLANE;                   // bwd flow ch1

    // CDNA5 WGP-scope prefetch (global_prefetch_b8, locality 3 -> near cache).
    // Flow is ~N(0,1): the bilinear taps span rows {y0, y0+1}, y0 in {y-1, y}
    // almost always. Warm rows y-1, y, y+1 of both dp_b planes at this thread's
    // column; each prefetched line covers both pixels of the pair.
    {
        const int ym = max(y - 1, 0), yp = min(y + 1, H_DIM - 1);   // uniform
        __builtin_prefetch(b0p + rem, 0, 3);
        __builtin_prefetch(b1p + rem, 0, 3);
        __builtin_prefetch(b0p + ym * W_DIM + x, 0, 3);
        __builtin_prefetch(b1p + ym * W_DIM + x, 0, 3);
        __builtin_prefetch(b0p + yp * W_DIM + x, 0, 3);
        __builtin_prefetch(b1p + yp * W_DIM + x, 0, 3);
    }

    // Stage the streaming (read-once) dp_f pair through LDS with the CDNA5
    // async data mover (b64 per plane). Each lane copies its own 8B; a
    // wave-local s_wait_asynccnt==0 makes the lane's own LDS slot visible,
    // so no workgroup barrier is required.
    __shared__ v2f sf0[BLOCK];
    __shared__ v2f sf1[BLOCK];
    {
        uint32_t l0 = (uint32_t)(uintptr_t)&sf0[tid];  // low 32 bits of a generic
        uint32_t l1 = (uint32_t)(uintptr_t)&sf1[tid];  // LDS pointer == DS offset
        const float* g0 = f0p + rem;
        const float* g1 = f1p + rem;
        asm volatile("global_load_async_to_lds_b64 %0, %1, off\n\t"
                     "global_load_async_to_lds_b64 %2, %3, off"
                     :: "v"(l0), "v"(g0), "v"(l1), "v"(g1)
                     : "memory");
#if __has_builtin(__builtin_amdgcn_s_wait_asynccnt)
        __builtin_amdgcn_s_wait_asynccnt(0);
#else
        asm volatile("s_wait_asynccnt 0x0" ::: "memory");
#endif
    }
    const v2f F0 = sf0[tid];   // channel-0 flow for pixels (x, x+1)
    const v2f F1 = sf1[tid];   // channel-1 flow for pixels (x, x+1)

    float r0a, r1a, occa, r0b, r1b, occb;
    sample_one(b0p, b1p, y, x,     F0.x, F1.x, r0a, r1a, occa);
    sample_one(b0p, b1p, y, x + 1, F0.y, F1.y, r0b, r1b, occb);

    // Outputs are write-once / never re-read: 8B non-temporal stores so the
    // 96 MB of results do not evict the gather-hot dp_b lines from L2
    // (dp_b = 64 MB fits comfortably in the 192 MB L2). All pairs 8B-aligned.
    const v2f occv = {occa, occb};
    const v2f r0v  = {r0a, r0b};
    const v2f r1v  = {r1a, r1b};
    const size_t rb = (size_t)b * 2 * PLANE + rem;
    __builtin_nontemporal_store(occv, (v2f*)(occ_out + idx0));
    __builtin_nontemporal_store(r0v,  (v2f*)(rev_out + rb));
    __builtin_nontemporal_store(r1v,  (v2f*)(rev_out + rb + PLANE));
}

extern "C" void kernel_launch(void* const* d_in, const int* in_sizes, int n_in,
                              void* d_out, int out_size, void* d_ws, size_t ws_size,
                              hipStream_t stream) {
    const float* dp_f = (const float*)d_in[0];
    const float* dp_b = (const float*)d_in[1];
    const int total = in_sizes[0];      // B*2*H*W
    const int npix  = total / 2;        // B*H*W  (8M: divisible by BLOCK*PIXPT)
    float* occ = (float*)d_out;         // first output: occlusion as 0/1 floats
    float* rev = (float*)d_out + npix;  // second output: dp_rev [B,2,H,W]
    const int blocks = (npix + BLOCK * PIXPT - 1) / (BLOCK * PIXPT);
    occlusion_kernel<<<blocks, BLOCK, 0, stream>>>(dp_f, dp_b, occ, rev, npix);
}